// GraphLearningModule_4209067950483
// MI455X (gfx1250) — compile-verified
//
#include <hip/hip_runtime.h>

// Problem constants (match reference).
#define B_   16
#define N_   50000
#define K_   32
#define E_   6
#define F_   3
#define EP1_ 7
#define BF_  (B_ * F_)   // 48 floats of f-features per node

// Kernel A tiling: 16 nodes per wave, 5 waves per block (160 threads).
// N/16 = 3125 tiles = 5^5  ->  exactly 625 blocks, no partial waves, so
// every wave reaching the WMMAs has EXEC == all-ones (ISA 7.12 requirement).
#define WAVES_A 5
#define TILES_  (N_ / 16)            // 3125
#define BLOCKS_A (TILES_ / WAVES_A)  // 625

typedef __attribute__((ext_vector_type(2))) float v2f;
typedef __attribute__((ext_vector_type(8))) float v8f;

// ---------------------------------------------------------------------------
// Kernel A: build f-table  ftab[n][b][fo]  (N x 16 x 3 f32, gather-friendly).
//
// g[n,fo] = sum_e emb[n,e]*W[fo,e+1] + bias[fo]   -> via V_WMMA_F32_16X16X4_F32
//           (16-node tile per wave, K padded 6->8, two k-steps, bias seeded
//            into the C accumulator per column)
// f[b,n,fo] = leakyrelu(x[b,n]*W[fo,0] + g[n,fo]) -> VALU expansion over b
//
// ISA 7.12.2 layouts used:
//   A 16x4 f32 : lane m=lane&15, khalf=lane>>4 ; VGPR0=A[m,2*khalf+0],
//                VGPR1=A[m,2*khalf+1]
//   B 4x16 f32 : col n=lane&15 ; VGPR0=B[2*khalf+0,n], VGPR1=B[2*khalf+1,n]
//   C 16x16 f32: VGPR v -> row (v + 8*khalf), col lane&15
// ---------------------------------------------------------------------------
__global__ __launch_bounds__(32 * WAVES_A) void gl_fc_wmma_kernel(
    const float* __restrict__ x,      // (B, N)
    const float* __restrict__ emb,    // (N, E)
    const float* __restrict__ W,      // (F, E+1) row-major
    const float* __restrict__ bias,   // (F)
    float* __restrict__ ftab)         // (N, B, F)
{
  __shared__ float g_lds[WAVES_A][16 * F_];   // per-wave 16x3 g-tile

  const int wave  = threadIdx.x >> 5;
  const int lane  = threadIdx.x & 31;
  const int tile  = blockIdx.x * WAVES_A + wave;   // always < TILES_
  const int node0 = tile * 16;
  const int m     = lane & 15;
  const int khalf = lane >> 4;                     // 0 or 1

  // ---- A operands: emb rows for 16 nodes, K split {0..3},{4..7} (pad 6,7)
  const float* er = emb + (size_t)(node0 + m) * E_;
  v2f a0, a1, b0, b1;
  {
    const int k0 = khalf * 2;               // 0 or 2
    a0.x = er[k0];
    a0.y = er[k0 + 1];
    const int k1 = 4 + khalf * 2;           // 4 or 6
    a1.x = (k1     < E_) ? er[k1]     : 0.0f;
    a1.y = (k1 + 1 < E_) ? er[k1 + 1] : 0.0f;
  }
  // ---- B operands: B[k, n] = W[n, k+1] for n<3, K padded with zeros
  {
    const bool col = (m < F_);
    const int k0 = khalf * 2;
    b0.x = col ? W[m * EP1_ + k0 + 1] : 0.0f;
    b0.y = col ? W[m * EP1_ + k0 + 2] : 0.0f;
    const int k1 = 4 + khalf * 2;
    b1.x = (col && (k1     < E_)) ? W[m * EP1_ + k1 + 1] : 0.0f;
    b1.y = (col && (k1 + 1 < E_)) ? W[m * EP1_ + k1 + 2] : 0.0f;
  }
  // ---- C seeded with bias per output column
  v8f c;
  const float binit = (m < F_) ? bias[m] : 0.0f;
#pragma unroll
  for (int i = 0; i < 8; ++i) c[i] = binit;

  c = __builtin_amdgcn_wmma_f32_16x16x4_f32(false, a0, false, b0,
                                            (short)0, c, false, false);
  c = __builtin_amdgcn_wmma_f32_16x16x4_f32(false, a1, false, b1,
                                            (short)0, c, false, false);

  // ---- spill useful 16x3 block of C to LDS (cols 0..2 only)
  if (m < F_) {
#pragma unroll
    for (int v = 0; v < 8; ++v) {
      g_lds[wave][(v + khalf * 8) * F_ + m] = c[v];
    }
  }
  __syncthreads();

  {
    const float w0 = W[0 * EP1_];
    const float w1 = W[1 * EP1_];
    const float w2 = W[2 * EP1_];
    // 256 (b, node) jobs per wave, 8 per lane; lanes 0..15 share b for
    // coalesced x loads.
#pragma unroll
    for (int i = 0; i < 8; ++i) {
      const int job = i * 32 + lane;          // 0..255
      const int b   = job >> 4;               // 0..15
      const int mm  = job & 15;               // node within tile
      const float xv = x[(size_t)b * N_ + (node0 + mm)];
      float f0 = fmaf(xv, w0, g_lds[wave][mm * F_ + 0]);
      float f1 = fmaf(xv, w1, g_lds[wave][mm * F_ + 1]);
      float f2 = fmaf(xv, w2, g_lds[wave][mm * F_ + 2]);
      f0 = (f0 >= 0.0f) ? f0 : 0.2f * f0;     // leaky relu 0.2
      f1 = (f1 >= 0.0f) ? f1 : 0.2f * f1;
      f2 = (f2 >= 0.0f) ? f2 : 0.2f * f2;
      float* dst = ftab + (size_t)(node0 + mm) * BF_ + b * F_;
      dst[0] = f0; dst[1] = f1; dst[2] = f2;
    }
  }
}

// ---------------------------------------------------------------------------
// Kernel B: one wave per node, lane <-> neighbor slot k.
//   w[n,k] = (1/16) * sum_b exp2( coef * ||f[n,b,:] - f[nb,b,:]||^2 ),
//   coef = -log2(e)/(2*theta);  masked to 0 where neighbor == -1.
// Self features: wave-uniform broadcast loads; neighbor features: 12x b128.
// All traffic stays in the 192MB L2 (f-table is 9.6MB).
// ---------------------------------------------------------------------------
__global__ __launch_bounds__(256) void gl_edge_weight_kernel(
    const float* __restrict__ ftab,     // (N, B*F)
    const int*   __restrict__ nbr,      // (N, K) int32
    const float* __restrict__ theta_p,  // scalar
    float* __restrict__ out)            // (N, K)
{
  const int wave = threadIdx.x >> 5;
  const int lane = threadIdx.x & 31;
  const int n    = blockIdx.x * 8 + wave;   // grid = N/8 exactly -> n < N_

  // Self features (uniform across lanes -> broadcast).
  float fs[BF_];
  const float4* ps = (const float4*)(ftab + (size_t)n * BF_);
#pragma unroll
  for (int i = 0; i < BF_ / 4; ++i) {
    const float4 q = ps[i];
    fs[i * 4 + 0] = q.x; fs[i * 4 + 1] = q.y;
    fs[i * 4 + 2] = q.z; fs[i * 4 + 3] = q.w;
  }

  const float theta = theta_p[0];
  const float coef  = -1.4426950408889634f / (2.0f * theta);  // -log2e/(2θ)

  const int nb = nbr[(size_t)n * K_ + lane];
  float w = 0.0f;
  if (nb >= 0) {
    const float4* pn = (const float4*)(ftab + (size_t)nb * BF_);
    float fl[BF_];
#pragma unroll
    for (int i = 0; i < BF_ / 4; ++i) {
      const float4 q = pn[i];
      fl[i * 4 + 0] = q.x; fl[i * 4 + 1] = q.y;
      fl[i * 4 + 2] = q.z; fl[i * 4 + 3] = q.w;
    }
    float acc = 0.0f;
#pragma unroll
    for (int b = 0; b < B_; ++b) {
      const float d0 = fs[b * F_ + 0] - fl[b * F_ + 0];
      const float d1 = fs[b * F_ + 1] - fl[b * F_ + 1];
      const float d2 = fs[b * F_ + 2] - fl[b * F_ + 2];
      const float s  = fmaf(d0, d0, fmaf(d1, d1, d2 * d2));
      acc += __builtin_amdgcn_exp2f(coef * s);   // v_exp_f32
    }
    w = acc * (1.0f / (float)B_);
  }
  out[(size_t)n * K_ + lane] = w;
}

// ---------------------------------------------------------------------------
extern "C" void kernel_launch(void* const* d_in, const int* in_sizes, int n_in,
                              void* d_out, int out_size, void* d_ws, size_t ws_size,
                              hipStream_t stream) {
  const float* x     = (const float*)d_in[0];   // (B,N)      f32
  const float* emb   = (const float*)d_in[1];   // (N,E)      f32
  const float* W     = (const float*)d_in[2];   // (F,E+1)    f32
  const float* bias  = (const float*)d_in[3];   // (F,)       f32
  const float* theta = (const float*)d_in[4];   // scalar     f32
  const int*   nbr   = (const int*)  d_in[5];   // (N,K)      i32
  float*       out   = (float*)d_out;           // (N,K)      f32
  float*       ftab  = (float*)d_ws;            // (N,B,F)    9.6 MB scratch

  // Kernel A: 3125 tiles = 625 blocks x 5 waves (exact; EXEC all-ones at WMMA).
  gl_fc_wmma_kernel<<<dim3(BLOCKS_A), dim3(32 * WAVES_A), 0, stream>>>(
      x, emb, W, bias, ftab);

  // Kernel B: one wave per node, 8 nodes per block; 50000/8 = 6250 exact.
  gl_edge_weight_kernel<<<dim3(N_ / 8), dim3(256), 0, stream>>>(
      ftab, nbr, theta, out);
}